// Net_77060303225524
// MI455X (gfx1250) — compile-verified
//
#include <hip/hip_runtime.h>
#include <math.h>

typedef __attribute__((ext_vector_type(2))) float v2f;
typedef __attribute__((ext_vector_type(8))) float v8f;

constexpr int NND   = 2048;            // nodes == image dim
constexpr long NSQ  = (long)NND * NND; // 4Mi elements (16 MB)
constexpr int NEDGE = 32768;
constexpr int FIN   = 1433;
constexpr int HID   = 64;
constexpr int NC    = 7;
constexpr int P1R   = 1024;
constexpr int P2R   = 512;

// ---------------------------------------------------------------- scatter A
__global__ __launch_bounds__(256) void scatter_edges(const int* __restrict__ ei,
                                                     float* __restrict__ A) {
  int e = blockIdx.x * 256 + threadIdx.x;
  if (e < NEDGE) {
    int s = ei[e];
    int d = ei[NEDGE + e];
    atomicAdd(&A[(long)s * NND + d], 1.0f);
  }
}

// --------------------------------------------- fused conv1+pool+conv2+pool
// meta's 3 input channels are identical (=A) -> conv1 weights summed over in.
// Block computes an 8x8 pooled2 tile for all 4 channels.
__global__ __launch_bounds__(256) void conv_front(const float* __restrict__ A,
                                                  const float* __restrict__ c1_w,
                                                  const float* __restrict__ c1_b,
                                                  const float* __restrict__ c2_w,
                                                  const float* __restrict__ c2_b,
                                                  float* __restrict__ pooled2) {
  __shared__ float sA[38][40];
  __shared__ float sP1[16][18][20];
  __shared__ float sW1[16][9];
  const int t = threadIdx.x;
  const int P2Y0 = blockIdx.y * 8, P2X0 = blockIdx.x * 8;
  const int a0y = 4 * P2Y0 - 3, a0x = 4 * P2X0 - 3;

  if (t < 144) {
    int c = t / 9, k = t % 9;
    sW1[c][k] = c1_w[c * 27 + k] + c1_w[c * 27 + 9 + k] + c1_w[c * 27 + 18 + k];
  }
  for (int idx = t; idx < 38 * 38; idx += 256) {
    int rr = idx / 38, cc = idx % 38;
    int gy = a0y + rr, gx = a0x + cc;
    float v = 0.f;
    if (gy >= 0 && gy < NND && gx >= 0 && gx < NND) v = A[(long)gy * NND + gx];
    sA[rr][cc] = v;
  }
  __syncthreads();

  // pooled1 = relu(maxpool(conv1)) on an 18x18 halo tile, 16 channels
  for (int idx = t; idx < 16 * 18 * 18; idx += 256) {
    int c = idx / 324, rem = idx % 324;
    int ry = rem / 18, rx = rem % 18;
    int gy = 2 * P2Y0 - 1 + ry, gx = 2 * P2X0 - 1 + rx;
    float out = 0.f;
    if (gy >= 0 && gy < P1R && gx >= 0 && gx < P1R) {
      float m = -3.4e38f;
#pragma unroll
      for (int dy = 0; dy < 2; ++dy)
#pragma unroll
        for (int dx = 0; dx < 2; ++dx) {
          float s = 0.f;
#pragma unroll
          for (int ky = 0; ky < 3; ++ky)
#pragma unroll
            for (int kx = 0; kx < 3; ++kx)
              s += sW1[c][ky * 3 + kx] * sA[2 * ry + dy + ky][2 * rx + dx + kx];
          m = fmaxf(m, s);
        }
      out = fmaxf(m + c1_b[c], 0.f);
    }
    sP1[c][ry][rx] = out;
  }
  __syncthreads();

  // pooled2 = relu(maxpool(conv2)) : one output per thread (4ch x 8 x 8)
  {
    int c2 = t >> 6, qy = (t >> 3) & 7, qx = t & 7;
    float a00 = 0.f, a01 = 0.f, a10 = 0.f, a11 = 0.f;
    for (int ci = 0; ci < 16; ++ci)
#pragma unroll
      for (int ky = 0; ky < 3; ++ky)
#pragma unroll
        for (int kx = 0; kx < 3; ++kx) {
          float w = c2_w[((c2 * 16 + ci) * 3 + ky) * 3 + kx];
          a00 += w * sP1[ci][2 * qy + ky][2 * qx + kx];
          a01 += w * sP1[ci][2 * qy + ky][2 * qx + 1 + kx];
          a10 += w * sP1[ci][2 * qy + 1 + ky][2 * qx + kx];
          a11 += w * sP1[ci][2 * qy + 1 + ky][2 * qx + 1 + kx];
        }
    float m = fmaxf(fmaxf(a00, a01), fmaxf(a10, a11));
    float out = fmaxf(m + c2_b[c2], 0.f);
    pooled2[((long)c2 * P2R + (P2Y0 + qy)) * P2R + (P2X0 + qx)] = out;
  }
}

// ------------------- fused tconv1+relu+tconv2+sigmoid, times A -> attended
// kernel=2 stride=2 tconvs: each output pixel depends on one input pixel.
__global__ __launch_bounds__(256) void tconv_fused(const float* __restrict__ pooled2,
                                                   const float* __restrict__ A,
                                                   const float* __restrict__ t1_w,
                                                   const float* __restrict__ t1_b,
                                                   const float* __restrict__ t2_w,
                                                   const float* __restrict__ t2_b,
                                                   float* __restrict__ attended) {
  __shared__ float sW1[256], sW2[192], sB1[16], sB2[3];
  int t = threadIdx.y * 16 + threadIdx.x;
  sW1[t] = t1_w[t];
  if (t < 192) sW2[t] = t2_w[t];
  if (t < 16) sB1[t] = t1_b[t];
  if (t < 3) sB2[t] = t2_b[t];
  __syncthreads();

  int x1 = blockIdx.x * 16 + threadIdx.x;  // t1-resolution coords [0,1024)
  int y1 = blockIdx.y * 16 + threadIdx.y;
  int yp = y1 >> 1, xp = x1 >> 1;
  float p[4];
#pragma unroll
  for (int ci = 0; ci < 4; ++ci) p[ci] = pooled2[((long)ci * P2R + yp) * P2R + xp];
  int kh = y1 & 1, kw = x1 & 1;
  float t1c[16];
#pragma unroll
  for (int c = 0; c < 16; ++c) {
    float s = sB1[c];
#pragma unroll
    for (int ci = 0; ci < 4; ++ci) s += p[ci] * sW1[((ci * 16 + c) * 2 + kh) * 2 + kw];
    t1c[c] = fmaxf(s, 0.f);
  }
#pragma unroll
  for (int dy = 0; dy < 2; ++dy)
#pragma unroll
    for (int dx = 0; dx < 2; ++dx) {
      int Y = 2 * y1 + dy, X = 2 * x1 + dx;
      float a = A[(long)Y * NND + X];
#pragma unroll
      for (int o = 0; o < 3; ++o) {
        float s = sB2[o];
#pragma unroll
        for (int c = 0; c < 16; ++c) s += t1c[c] * sW2[((c * 3 + o) * 2 + dy) * 2 + dx];
        float att = 1.f / (1.f + expf(-s));
        attended[((long)o * NND + Y) * NND + X] = a * att;
      }
    }
}

// ------------------------------------------------ f32 WMMA GEMM  C = A @ B
// wave: 64x32 C macro-tile (4x2 WMMA tiles), block: 8 waves -> 128x128 tile.
template <bool GUARD>
__global__ __launch_bounds__(256) void gemm_wmma(const float* __restrict__ Am,
                                                 const float* __restrict__ Bm,
                                                 float* __restrict__ Cm,
                                                 int M, int Ncols, int K,
                                                 const float* __restrict__ bias,
                                                 int relu) {
  const int lane = threadIdx.x & 31;
  const int wave = threadIdx.x >> 5;
  const int half = lane >> 4;
  const int l16 = lane & 15;
  const int row0 = blockIdx.y * 128 + (wave >> 2) * 64;
  const int col0 = blockIdx.x * 128 + (wave & 3) * 32;

  v8f acc[4][2];
#pragma unroll
  for (int i = 0; i < 4; ++i)
#pragma unroll
    for (int j = 0; j < 2; ++j) acc[i][j] = (v8f){0, 0, 0, 0, 0, 0, 0, 0};

  for (int k = 0; k < K; k += 4) {
    const int ka = k + 2 * half;           // A/B K index for vgpr0, ka+1 for vgpr1
    v2f af[4], bf[2];
#pragma unroll
    for (int i = 0; i < 4; ++i) {
      int m = row0 + i * 16 + l16;
      if (GUARD) {
        af[i].x = (m < M && ka < K) ? Am[(long)m * K + ka] : 0.f;
        af[i].y = (m < M && ka + 1 < K) ? Am[(long)m * K + ka + 1] : 0.f;
      } else {
        af[i].x = Am[(long)m * K + ka];
        af[i].y = Am[(long)m * K + ka + 1];
      }
    }
#pragma unroll
    for (int j = 0; j < 2; ++j) {
      int n = col0 + j * 16 + l16;
      if (GUARD) {
        bf[j].x = (n < Ncols && ka < K) ? Bm[(long)ka * Ncols + n] : 0.f;
        bf[j].y = (n < Ncols && ka + 1 < K) ? Bm[(long)(ka + 1) * Ncols + n] : 0.f;
      } else {
        bf[j].x = Bm[(long)ka * Ncols + n];
        bf[j].y = Bm[(long)(ka + 1) * Ncols + n];
      }
    }
#pragma unroll
    for (int i = 0; i < 4; ++i)
#pragma unroll
      for (int j = 0; j < 2; ++j)
        acc[i][j] = __builtin_amdgcn_wmma_f32_16x16x4_f32(
            false, af[i], false, bf[j], (short)0, acc[i][j], false, false);
  }

#pragma unroll
  for (int i = 0; i < 4; ++i)
#pragma unroll
    for (int j = 0; j < 2; ++j) {
      int n = col0 + j * 16 + l16;
#pragma unroll
      for (int v = 0; v < 8; ++v) {
        int m = row0 + i * 16 + v + 8 * half;
        if (!GUARD || (m < M && n < Ncols)) {
          float val = acc[i][j][v];
          if (bias) val += bias[n];
          if (relu) val = fmaxf(val, 0.f);
          Cm[(long)m * Ncols + n] = val;
        }
      }
    }
}

// ----------------------------------------------------------- small kernels
__global__ __launch_bounds__(256) void cbrt_inplace(float* __restrict__ P, long n) {
  for (long i = blockIdx.x * 256L + threadIdx.x; i < n; i += (long)gridDim.x * 256)
    P[i] = cbrtf(P[i]);
}

__global__ __launch_bounds__(256) void fix_diag(float* __restrict__ M) {
  int i = blockIdx.x * 256 + threadIdx.x;
  if (i < NND) {
    float d = M[(long)i * NND + i];
    if (!(d > 0.f)) M[(long)i * NND + i] = d + 1.0f;
  }
}

__global__ __launch_bounds__(256) void colsum_partial(const float* __restrict__ M,
                                                      float* __restrict__ deg) {
  int j = blockIdx.x * 256 + threadIdx.x;  // column
  int r0 = blockIdx.y * (NND / 16);
  float s = 0.f;
  for (int i = 0; i < NND / 16; ++i) s += M[(long)(r0 + i) * NND + j];
  atomicAdd(&deg[j], s);
}

__global__ __launch_bounds__(256) void make_dinv(const float* __restrict__ deg,
                                                 float* __restrict__ dinv) {
  int j = blockIdx.x * 256 + threadIdx.x;
  if (j < NND) {
    float d = deg[j];
    dinv[j] = d > 0.f ? rsqrtf(d) : 0.f;
  }
}

// AhatT[i][j] = dinv[i]*dinv[j]*M2[j][i]  (scaled LDS-tiled transpose)
__global__ __launch_bounds__(256) void ahatT_kernel(const float* __restrict__ M2,
                                                    const float* __restrict__ dinv,
                                                    float* __restrict__ AT) {
  __shared__ float tile[32][33];
  int bx = blockIdx.x * 32, by = blockIdx.y * 32;
  for (int r = threadIdx.y; r < 32; r += 8)
    tile[r][threadIdx.x] = M2[(long)(by + r) * NND + bx + threadIdx.x];
  __syncthreads();
  for (int r = threadIdx.y; r < 32; r += 8) {
    int i = bx + r, j = by + threadIdx.x;
    AT[(long)i * NND + j] = dinv[i] * dinv[j] * tile[threadIdx.x][r];
  }
}

__global__ __launch_bounds__(256) void log_softmax_rows(const float* __restrict__ O,
                                                        float* __restrict__ out) {
  int r = blockIdx.x * 256 + threadIdx.x;
  if (r >= NND) return;
  float v[NC], mx = -3.4e38f;
#pragma unroll
  for (int c = 0; c < NC; ++c) {
    v[c] = O[(long)r * NC + c];
    mx = fmaxf(mx, v[c]);
  }
  float s = 0.f;
#pragma unroll
  for (int c = 0; c < NC; ++c) s += expf(v[c] - mx);
  float lse = mx + logf(s);
#pragma unroll
  for (int c = 0; c < NC; ++c) out[(long)r * NC + c] = v[c] - lse;
}

// ------------------------------------------------------------------ driver
extern "C" void kernel_launch(void* const* d_in, const int* in_sizes, int n_in,
                              void* d_out, int out_size, void* d_ws, size_t ws_size,
                              hipStream_t stream) {
  const float* x    = (const float*)d_in[0];
  const int*   ei   = (const int*)d_in[1];
  const float* W1   = (const float*)d_in[2];
  const float* b1   = (const float*)d_in[3];
  const float* W2   = (const float*)d_in[4];
  const float* b2   = (const float*)d_in[5];
  const float* c1_w = (const float*)d_in[6];
  const float* c1_b = (const float*)d_in[7];
  const float* c2_w = (const float*)d_in[8];
  const float* c2_b = (const float*)d_in[9];
  const float* t1_w = (const float*)d_in[10];
  const float* t1_b = (const float*)d_in[11];
  const float* t2_w = (const float*)d_in[12];
  const float* t2_b = (const float*)d_in[13];
  float* out = (float*)d_out;

  // workspace layout (floats); regions reused once their producer is dead
  float* wsA   = (float*)d_ws;                 // A; later tmp1
  float* wsP   = wsA + NSQ;                    // P -> M2 (in place)
  float* wsP2  = wsP + NSQ;                    // pooled2 (4*512*512)
  float* wsATT = wsP2 + (long)4 * P2R * P2R;   // attended (3*NSQ); later AhatT+misc
  float* AhatT = wsATT;                        // reuse (attended dead after P)
  float* dinv  = wsATT + NSQ;
  float* deg   = dinv + NND;
  float* XW    = deg + NND;                    // 2048x64
  float* Hbuf  = XW + (long)NND * HID;         // 2048x64
  float* HW2   = Hbuf + (long)NND * HID;       // 2048x7 (reserve 2048x8)
  float* Obuf  = HW2 + (long)NND * 8;          // 2048x7 (reserve 2048x8)
  size_t need = (size_t)(5 * NSQ + (long)4 * P2R * P2R) * sizeof(float);
  if (ws_size < need) return;

  // 1) A = scatter-add(edges)
  hipMemsetAsync(wsA, 0, NSQ * sizeof(float), stream);
  scatter_edges<<<NEDGE / 256, 256, 0, stream>>>(ei, wsA);

  // 2) conv1+relu+pool+conv2+relu+pool (fused, LDS-tiled)
  conv_front<<<dim3(P2R / 8, P2R / 8), 256, 0, stream>>>(wsA, c1_w, c1_b, c2_w, c2_b, wsP2);

  // 3) tconv1+relu+tconv2+sigmoid, attended = A * att (fused)
  tconv_fused<<<dim3(P1R / 16, P1R / 16), dim3(16, 16), 0, stream>>>(
      wsP2, wsA, t1_w, t1_b, t2_w, t2_b, wsATT);

  const float* att0 = wsATT;
  const float* att1 = wsATT + NSQ;
  const float* att2 = wsATT + 2 * NSQ;

  // 4) tmp1 = att0 @ att1   (A region dead -> tmp1)
  float* tmp1 = wsA;
  gemm_wmma<false><<<dim3(NND / 128, NND / 128), 256, 0, stream>>>(
      att0, att1, tmp1, NND, NND, NND, nullptr, 0);
  // 5) P = tmp1 @ att2
  gemm_wmma<false><<<dim3(NND / 128, NND / 128), 256, 0, stream>>>(
      tmp1, att2, wsP, NND, NND, NND, nullptr, 0);

  // 6) M2 = cbrt(P) in place; 7) add missing self-loops on the diagonal
  cbrt_inplace<<<4096, 256, 0, stream>>>(wsP, NSQ);
  fix_diag<<<NND / 256, 256, 0, stream>>>(wsP);

  // 8) deg = column sums, dinv = rsqrt(deg)
  hipMemsetAsync(deg, 0, NND * sizeof(float), stream);
  colsum_partial<<<dim3(NND / 256, 16), 256, 0, stream>>>(wsP, deg);
  make_dinv<<<NND / 256, 256, 0, stream>>>(deg, dinv);

  // 9) AhatT[i][j] = dinv[i]*dinv[j]*M2[j][i]  (attended region now dead)
  ahatT_kernel<<<dim3(NND / 32, NND / 32), dim3(32, 8), 0, stream>>>(wsP, dinv, AhatT);

  // 10) XW = x @ W1   (K=1433 ragged -> guarded)
  gemm_wmma<true><<<dim3(1, NND / 128), 256, 0, stream>>>(
      x, W1, XW, NND, HID, FIN, nullptr, 0);
  // 11) H = relu(AhatT @ XW + b1)
  gemm_wmma<true><<<dim3(1, NND / 128), 256, 0, stream>>>(
      AhatT, XW, Hbuf, NND, HID, NND, b1, 1);
  // 12) HW2 = H @ W2
  gemm_wmma<true><<<dim3(1, NND / 128), 256, 0, stream>>>(
      Hbuf, W2, HW2, NND, NC, HID, nullptr, 0);
  // 13) O = AhatT @ HW2 + b2
  gemm_wmma<true><<<dim3(1, NND / 128), 256, 0, stream>>>(
      AhatT, HW2, Obuf, NND, NC, NND, b2, 0);

  // 14) log_softmax rows -> output
  log_softmax_rows<<<NND / 256, 256, 0, stream>>>(Obuf, out);
}